// NLRidge_11029476016196
// MI455X (gfx1250) — compile-verified
//
#include <hip/hip_runtime.h>

// ---------------------------------------------------------------------------
// NLRidge (BM3D-style) denoiser for MI455X / gfx1250.
//  - fp32 WMMA (v_wmma_f32_16x16x4_f32) for Q = Y*Y^T and X_hat = theta*Y
//  - one workgroup per patch group, all intermediates in LDS
//  - aggregation via float atomics directly into folded num/den images
// ---------------------------------------------------------------------------

namespace {

constexpr int   H_    = 320;
constexpr int   W_    = 320;
constexpr int   P_    = 7;
constexpr int   PSQ   = P_ * P_;          // 49
constexpr int   nH    = H_ - P_ + 1;      // 314
constexpr int   nW    = W_ - P_ + 1;      // 314
constexpr int   NPATCH= nH * nW;          // 98596
constexpr int   STEPP = 4;
constexpr int   Rh    = 80;               // _ref_pos(314,4): 0,4,...,312,313
constexpr int   Rw    = 80;
constexpr int   NGRP  = Rh * Rw;          // 6400
constexpr int   WIN_  = 37;
constexpr int   V_    = 18;               // WIN/2
constexpr int   NCAND = WIN_ * WIN_;      // 1369
constexpr int   XWDIM = 2 * V_ + P_;      // 43  (window span touched by candidates)
constexpr int   K1_   = 18;
constexpr int   K2_   = 55;
constexpr float DVAL  = 25.0f * 25.0f * (float)PSQ;   // sigma^2 * n = 30625
constexpr int   IMGSZ = H_ * W_;          // 102400
constexpr int   NT_   = 256;              // threads per block
constexpr int   NWAVE = NT_ / 32;

// workspace layout (float units)
constexpr size_t OFF_N2   = 0;                       // NPATCH (98596) pad->98816
constexpr size_t OFF_IDX  = 98816;                   // NGRP*K2_ ints = 352000
constexpr size_t OFF_NUM  = OFF_IDX + 352000;        // IMGSZ
constexpr size_t OFF_DEN  = OFF_NUM + IMGSZ;         // IMGSZ
constexpr size_t OFF_DEN1 = OFF_DEN + IMGSZ;         // IMGSZ

typedef __attribute__((ext_vector_type(2))) float v2f;
typedef __attribute__((ext_vector_type(8))) float v8f;

} // namespace

// ---------------------------------------------------------------------------
__global__ void zero_kernel(float* __restrict__ p, int n) {
    int t = blockIdx.x * blockDim.x + threadIdx.x;
    if (t < n) p[t] = 0.0f;
}

// n2[q] = ||patch(q)||^2 / 49  over valid patch positions
__global__ void patch_norm_kernel(const float* __restrict__ img,
                                  float* __restrict__ n2) {
    int t = blockIdx.x * blockDim.x + threadIdx.x;
    if (t >= NPATCH) return;
    int qh = t / nW, qw = t - qh * nW;
    float s = 0.0f;
    for (int a = 0; a < P_; ++a)
        for (int b = 0; b < P_; ++b) {
            float x = img[(qh + a) * W_ + (qw + b)];
            s += x * x;
        }
    n2[t] = s * (1.0f / (float)PSQ);
}

// ---------------------------------------------------------------------------
// Block matching: one workgroup per reference position; top-KG by iterative
// argmin with smaller-index tie-break. Center displacement forced to -inf,
// out-of-range candidates +inf (reference pads norms with inf).
template <int KG>
__global__ void block_match_kernel(const float* __restrict__ img,
                                   const float* __restrict__ n2,
                                   int* __restrict__ outIdx) {
    const int gw = blockIdx.x, gh = blockIdx.y;
    const int refh = (gh < Rh - 1) ? gh * STEPP : (nH - 1);
    const int refw = (gw < Rw - 1) ? gw * STEPP : (nW - 1);
    const int tid  = threadIdx.x;
    const float INF = __builtin_inff();

    __shared__ float xw[XWDIM][XWDIM + 1];
    __shared__ float dsh[NCAND];
    __shared__ float rv[NT_];
    __shared__ int   ri[NT_];

    // stage window (clamped reads; clamped values only feed OOB candidates)
    for (int t = tid; t < XWDIM * XWDIM; t += NT_) {
        int r = t / XWDIM, c = t - r * XWDIM;
        int rr = refh - V_ + r, cc = refw - V_ + c;
        float v = 0.0f;
        if (rr >= 0 && rr < H_ && cc >= 0 && cc < W_) v = img[rr * W_ + cc];
        xw[r][c] = v;
    }
    __syncthreads();

    for (int d = tid; d < NCAND; d += NT_) {
        int dh = d / WIN_, dw = d - dh * WIN_;
        int ch = refh + dh - V_, cw = refw + dw - V_;
        float dist;
        if (dh == V_ && dw == V_) {
            dist = -INF;
        } else if (ch < 0 || ch >= nH || cw < 0 || cw >= nW) {
            dist = INF;
        } else {
            float corr = 0.0f;
            for (int a = 0; a < P_; ++a)
                for (int b = 0; b < P_; ++b)
                    corr += xw[V_ + a][V_ + b] * xw[dh + a][dw + b];
            dist = n2[ch * nW + cw] - (2.0f / (float)PSQ) * corr;
        }
        dsh[d] = dist;
    }
    __syncthreads();

    for (int sel = 0; sel < KG; ++sel) {
        float bv = INF; int bi = NCAND;
        for (int d = tid; d < NCAND; d += NT_) {
            float v = dsh[d];
            if (v < bv || (v == bv && d < bi)) { bv = v; bi = d; }
        }
        rv[tid] = bv; ri[tid] = bi;
        __syncthreads();
        for (int s = NT_ / 2; s > 0; s >>= 1) {
            if (tid < s) {
                float ov = rv[tid + s]; int oi = ri[tid + s];
                if (ov < rv[tid] || (ov == rv[tid] && oi < ri[tid])) {
                    rv[tid] = ov; ri[tid] = oi;
                }
            }
            __syncthreads();
        }
        if (tid == 0) {
            int d  = ri[0];
            int ch = refh + d / WIN_ - V_;
            int cw = refw + d % WIN_ - V_;
            outIdx[(gh * Rw + gw) * KG + sel] = ch * nW + cw;
            dsh[d] = INF;
        }
        __syncthreads();
    }
}

// ---------------------------------------------------------------------------
template <int KG, int KM>
__device__ inline void gather_group(const float* __restrict__ img,
                                    const int* psh,
                                    float (*Ysh)[64], int tid) {
    for (int t = tid; t < KM * 64; t += NT_) {
        int i = t >> 6, c = t & 63;
        float v = 0.0f;
        if (i < KG && c < PSQ) {
            int q  = psh[i];
            int qh = q / nW, qw = q - qh * nW;
            v = img[(qh + c / P_) * W_ + (qw + c % P_)];
        }
        Ysh[i][c] = v;
    }
}

// One workgroup per group:
//   gather -> Q (WMMA) -> Cholesky -> Qinv -> theta -> wts -> X_hat (WMMA)
//   -> atomic scatter into folded num/den images.
template <int KG, bool STEP2>
__global__ void group_denoise_kernel(const float* __restrict__ yImg,
                                     const float* __restrict__ xImg,
                                     const int* __restrict__ idxArr,
                                     float* __restrict__ num,
                                     float* __restrict__ den) {
    constexpr int KM = (KG + 15) & ~15;     // 18->32, 55->64
    constexpr int MT = KM / 16;             // M tiles

    __shared__ float Ysh[KM][64];           // group patches (rows>=KG zero)
    __shared__ float Qsh[KM][KM];           // Q -> L -> theta
    __shared__ float Zsh[KM][64];           // Z -> Qinv -> X_hat
    __shared__ float wsh[KM];
    __shared__ int   psh[KG];

    const int g    = blockIdx.x;
    const int tid  = threadIdx.x;
    const int lane = tid & 31;
    const int wave = tid >> 5;
    const int lm   = lane & 15;
    const int half = lane >> 4;

    if (tid < KG) psh[tid] = idxArr[g * KG + tid];
    __syncthreads();

    // ---- build Q from the "structure" image (X in step2, Y in step1) ----
    gather_group<KG, KM>(STEP2 ? xImg : yImg, psh, Ysh, tid);
    __syncthreads();

    // Q = S * S^T  via v_wmma_f32_16x16x4_f32 ; K = 49 padded to 52
    for (int tile = wave; tile < MT * MT; tile += NWAVE) {
        int tm = (tile / MT) * 16, tn = (tile % MT) * 16;
        v8f acc = {};
        for (int kb = 0; kb < 52; kb += 4) {
            v2f a, b;
            a.x = Ysh[tm + lm][kb + 2 * half];
            a.y = Ysh[tm + lm][kb + 2 * half + 1];
            b.x = Ysh[tn + lm][kb + 2 * half];
            b.y = Ysh[tn + lm][kb + 2 * half + 1];
            acc = __builtin_amdgcn_wmma_f32_16x16x4_f32(
                false, a, false, b, (short)0, acc, false, false);
        }
#pragma unroll
        for (int v = 0; v < 8; ++v)
            Qsh[tm + v + 8 * half][tn + lm] = acc[v];
    }
    __syncthreads();

    if (STEP2) {
        if (tid < KG) Qsh[tid][tid] += DVAL;     // Q = X X^T + Dval I
        __syncthreads();
        gather_group<KG, KM>(yImg, psh, Ysh, tid);   // now load Y (denoise target)
    }
    __syncthreads();

    // ---- cooperative Cholesky (lower triangle of Qsh becomes L) ----
    for (int j = 0; j < KG; ++j) {
        if (tid == 0) Qsh[j][j] = sqrtf(Qsh[j][j]);
        __syncthreads();
        float invd = 1.0f / Qsh[j][j];
        for (int i = j + 1 + tid; i < KG; i += NT_) Qsh[i][j] *= invd;
        __syncthreads();
        int m = KG - 1 - j;
        for (int t = tid; t < m * m; t += NT_) {
            int i = j + 1 + t / m, l = j + 1 + t % m;
            if (l <= i) Qsh[i][l] -= Qsh[i][j] * Qsh[l][j];
        }
        __syncthreads();
    }

    // ---- Qinv via two triangular solves, one column per thread ----
    if (tid < KG) {
        const int c = tid;
        for (int i = 0; i < KG; ++i) {             // L z = e_c
            float s = (i == c) ? 1.0f : 0.0f;
            for (int j = 0; j < i; ++j) s -= Qsh[i][j] * Zsh[j][c];
            Zsh[i][c] = s / Qsh[i][i];
        }
        for (int i = KG - 1; i >= 0; --i) {        // L^T q = z  (in place)
            float s = Zsh[i][c];
            for (int j = i + 1; j < KG; ++j) s -= Qsh[j][i] * Zsh[j][c];
            Zsh[i][c] = s / Qsh[i][i];
        }
    }
    __syncthreads();

    // ---- theta = I - Dval * Qinv (symmetric); overwrite Qsh, zero padding ----
    for (int t = tid; t < KM * KM; t += NT_) {
        int i = t / KM, j = t - i * KM;
        float v = 0.0f;
        if (i < KG && j < KG)
            v = ((i == j) ? 1.0f : 0.0f) - DVAL * Zsh[i][j];
        Qsh[i][j] = v;
    }
    __syncthreads();

    // wts_i = 1 / clip(sum_j theta[i][j]^2, 1/KG, 1)
    if (tid < KG) {
        float s = 0.0f;
        for (int j = 0; j < KG; ++j) s += Qsh[tid][j] * Qsh[tid][j];
        s = fminf(fmaxf(s, 1.0f / (float)KG), 1.0f);
        wsh[tid] = 1.0f / s;
    }
    __syncthreads();

    // ---- X_hat = theta * Y  via WMMA ; K = KM (theta padded with zeros) ----
    for (int tile = wave; tile < MT * 4; tile += NWAVE) {
        int tm = (tile / 4) * 16, tn = (tile % 4) * 16;
        v8f acc = {};
        for (int kb = 0; kb < KM; kb += 4) {
            v2f a, b;
            a.x = Qsh[tm + lm][kb + 2 * half];
            a.y = Qsh[tm + lm][kb + 2 * half + 1];
            b.x = Ysh[kb + 2 * half][tn + lm];
            b.y = Ysh[kb + 2 * half + 1][tn + lm];
            acc = __builtin_amdgcn_wmma_f32_16x16x4_f32(
                false, a, false, b, (short)0, acc, false, false);
        }
#pragma unroll
        for (int v = 0; v < 8; ++v)
            Zsh[tm + v + 8 * half][tn + lm] = acc[v];
    }
    __syncthreads();

    // ---- scatter: num += X_hat * w, den += w, folded directly into image ----
    for (int t = tid; t < KG * PSQ; t += NT_) {
        int i = t / PSQ, c = t - i * PSQ;
        int q = psh[i];
        int qh = q / nW, qw = q - qh * nW;
        int pix = (qh + c / P_) * W_ + (qw + c % P_);
        float w = wsh[i];
        atomicAdd(&num[pix], Zsh[i][c] * w);
        atomicAdd(&den[pix], w);
    }
}

// ---------------------------------------------------------------------------
__global__ void finalize_kernel(const float* __restrict__ num,
                                const float* __restrict__ den,
                                float* __restrict__ out) {
    int t = blockIdx.x * blockDim.x + threadIdx.x;
    if (t < IMGSZ) out[t] = num[t] / den[t];
}

// ---------------------------------------------------------------------------
extern "C" void kernel_launch(void* const* d_in, const int* in_sizes, int n_in,
                              void* d_out, int out_size, void* d_ws, size_t ws_size,
                              hipStream_t stream) {
    (void)in_sizes; (void)n_in; (void)out_size; (void)ws_size;
    const float* y = (const float*)d_in[0];
    float* ws   = (float*)d_ws;
    float* n2   = ws + OFF_N2;
    int*   idxb = (int*)(ws + OFF_IDX);
    float* num  = ws + OFF_NUM;
    float* den  = ws + OFF_DEN;
    float* den1 = ws + OFF_DEN1;
    float* out  = (float*)d_out;

    dim3 blk(NT_);
    dim3 grpGrid(NGRP);
    dim3 bmGrid(Rw, Rh);
    int zN = 2 * IMGSZ;                       // num & den are contiguous

    // ---------------- step 1 (k = 18) ----------------
    zero_kernel<<<(zN + NT_ - 1) / NT_, blk, 0, stream>>>(num, zN);
    patch_norm_kernel<<<(NPATCH + NT_ - 1) / NT_, blk, 0, stream>>>(y, n2);
    block_match_kernel<K1_><<<bmGrid, blk, 0, stream>>>(y, n2, idxb);
    group_denoise_kernel<K1_, false><<<grpGrid, blk, 0, stream>>>(y, nullptr, idxb, num, den);
    finalize_kernel<<<(IMGSZ + NT_ - 1) / NT_, blk, 0, stream>>>(num, den, den1);

    // ---------------- step 2 (k = 55) ----------------
    zero_kernel<<<(zN + NT_ - 1) / NT_, blk, 0, stream>>>(num, zN);
    patch_norm_kernel<<<(NPATCH + NT_ - 1) / NT_, blk, 0, stream>>>(den1, n2);
    block_match_kernel<K2_><<<bmGrid, blk, 0, stream>>>(den1, n2, idxb);
    group_denoise_kernel<K2_, true><<<grpGrid, blk, 0, stream>>>(y, den1, idxb, num, den);
    finalize_kernel<<<(IMGSZ + NT_ - 1) / NT_, blk, 0, stream>>>(num, den, out);
}